// DetectionConfidenceMap2keypoint_3kp_43602507989176
// MI455X (gfx1250) — compile-verified
//
#include <hip/hip_runtime.h>
#include <hip/hip_bf16.h>
#include <math.h>

// CDNA5 / gfx1250, wave32.
typedef float v2f __attribute__((ext_vector_type(2)));
typedef float v8f __attribute__((ext_vector_type(8)));

#define HH 96
#define WW 96
#define HWSZ (HH * WW)           // 9216 floats = 36 KB per heatmap
#define NMAPS 1024               // B*K = 16*64
#define N1 ((size_t)NMAPS * HWSZ)

// One block == one 96x96 heatmap. 256 threads = 8 wave32s.
// Phase 1: coalesced float4 sigmoid -> LDS + global Dk.
// Phase 2: waves 0..5 each own a 16-row strip; 24x V_WMMA_F32_16X16X4_F32
//          computes [rowsum, rowsum*x] per row (B col0 = ones, col1 = x).
// Phase 3: lane-level extract (C layout: lanes 0/16 hold col N=0, lanes 1/17
//          hold N=1), LDS reduce, thread 0 does soft-argmax + gather + keypoints.
__global__ __launch_bounds__(256) void fused_decode_kernel(
    const float* __restrict__ Rk,
    const float* __restrict__ tfRk,
    float* __restrict__ out)
{
    __shared__ __align__(16) float lds[HWSZ];
    __shared__ float redS[6][2];
    __shared__ float redSx[6][2];
    __shared__ float redSy[6][2];

    const int map   = blockIdx.x;            // 0..2047
    const int is_tf = (map >= NMAPS) ? 1 : 0;
    const int bk    = is_tf ? (map - NMAPS) : map;   // b*64 + k
    const int b     = bk >> 6;
    const int k     = bk & 63;

    const float* __restrict__ src = (is_tf ? tfRk : Rk) + (size_t)bk * HWSZ;
    float* __restrict__ dDk  = out + (is_tf ? N1 : 0) + (size_t)bk * HWSZ;
    float* __restrict__ kpO  = out + 2 * N1 + (is_tf ? 6144 : 0) + (size_t)b * 384;
    float* __restrict__ zO   = out + 2 * N1 + 12288 + (is_tf ? 1024 : 0) + (size_t)b * 64 + k;

    const int tid = threadIdx.x;

    // ---- Phase 1: sigmoid, fully coalesced b128 loads/stores -------------
    {
        const float4* __restrict__ s4 = (const float4*)src;
        float4* __restrict__ d4 = (float4*)dDk;
        float4* __restrict__ l4 = (float4*)lds;
        #pragma unroll
        for (int i = 0; i < 9; ++i) {              // 2304 float4 / 256 threads
            const int idx = tid + i * 256;
            float4 v = s4[idx];
            v.x = 1.0f / (1.0f + __expf(-v.x));
            v.y = 1.0f / (1.0f + __expf(-v.y));
            v.z = 1.0f / (1.0f + __expf(-v.z));
            v.w = 1.0f / (1.0f + __expf(-v.w));
            l4[idx] = v;
            d4[idx] = v;
        }
    }
    __syncthreads();

    // ---- Phase 2: WMMA row reductions (waves 0..5, EXEC all ones) --------
    const int wave = tid >> 5;
    const int lane = tid & 31;
    if (wave < 6) {
        const int r0   = wave << 4;                 // strip base row
        const int row  = r0 + (lane & 15);          // A: M = lane (mod 16)
        const int koff = (lane < 16) ? 0 : 2;       // A: K pair per half-wave
        const int n    = lane & 15;                 // B/C: N = lane (mod 16)
        const float sel0 = (n == 0) ? 1.0f : 0.0f;  // B col 0 -> ones
        const float sel1 = (n == 1) ? 1.0f : 0.0f;  // B col 1 -> x coords
        const float* __restrict__ rowp = lds + row * WW + koff;

        v8f c = {};
        #pragma unroll
        for (int c0 = 0; c0 < WW; c0 += 4) {
            v2f a = *(const v2f*)(rowp + c0);       // ds_load_b64
            v2f bb;
            bb.x = sel0 + sel1 * (float)(c0 + koff);
            bb.y = sel0 + sel1 * (float)(c0 + koff + 1);
            // D = A(16x4,f32) * B(4x16,f32) + C(16x16,f32)
            c = __builtin_amdgcn_wmma_f32_16x16x4_f32(
                    false, a, false, bb, (short)0, c, false, false);
        }

        // C layout: VGPR j -> lanes 0-15: M=j, N=lane; lanes 16-31: M=j+8.
        const int half = (lane >= 16) ? 1 : 0;
        if (n == 0) {            // lanes 0,16: rowsums for 8 rows each
            float s = 0.0f, sy = 0.0f;
            #pragma unroll
            for (int j = 0; j < 8; ++j) {
                const float r = c[j];
                s  += r;
                sy += r * (float)(r0 + half * 8 + j);
            }
            redS[wave][half]  = s;
            redSy[wave][half] = sy;
        } else if (n == 1) {     // lanes 1,17: x-moments for 8 rows each
            float sx = 0.0f;
            #pragma unroll
            for (int j = 0; j < 8; ++j) sx += c[j];
            redSx[wave][half] = sx;
        }
    }
    __syncthreads();

    // ---- Phase 3: epilogue (soft-argmax, gather from LDS, keypoints) -----
    if (tid == 0) {
        float zeta = 0.0f, kx = 0.0f, ky = 0.0f;
        #pragma unroll
        for (int w = 0; w < 6; ++w) {
            zeta += redS[w][0]  + redS[w][1];
            kx   += redSx[w][0] + redSx[w][1];
            ky   += redSy[w][0] + redSy[w][1];
        }
        const float kpx = rintf(kx / zeta);   // RNE == jnp.round
        const float kpy = rintf(ky / zeta);
        int wi = (int)kpx; wi = wi < 0 ? 0 : (wi > WW - 1 ? WW - 1 : wi);
        int hi = (int)kpy; hi = hi < 0 ? 0 : (hi > HH - 1 ? HH - 1 : hi);
        const float d = lds[hi * WW + wi];
        const float kp1x = truncf(kpx + kpx * d), kp1y = truncf(kpy + kpy * d);
        const float kp2x = truncf(kpx - kpx * d), kp2y = truncf(kpy - kpy * d);
        // keypoint layout [B, 3K, 2]
        kpO[k * 2 + 0]         = kpx;  kpO[k * 2 + 1]         = kpy;
        kpO[(64 + k) * 2 + 0]  = kp1x; kpO[(64 + k) * 2 + 1]  = kp1y;
        kpO[(128 + k) * 2 + 0] = kp2x; kpO[(128 + k) * 2 + 1] = kp2y;
        *zO = zeta;
    }
}

extern "C" void kernel_launch(void* const* d_in, const int* in_sizes, int n_in,
                              void* d_out, int out_size, void* d_ws, size_t ws_size,
                              hipStream_t stream) {
    (void)in_sizes; (void)n_in; (void)d_ws; (void)ws_size; (void)out_size;
    const float* Rk   = (const float*)d_in[0];
    const float* tfRk = (const float*)d_in[1];
    // d_in[2], d_in[3] are my_height/my_width == 96 (compile-time constants here)
    float* out = (float*)d_out;
    fused_decode_kernel<<<2 * NMAPS, 256, 0, stream>>>(Rk, tfRk, out);
}